// RNNLikeISTA_26860725469709
// MI455X (gfx1250) — compile-verified
//
#include <hip/hip_runtime.h>

// ============================================================================
// ISTA deconvolution + MLP head, fully fused for MI455X (gfx1250).
//
// Strategy:
//  * One workgroup per 16 batch rows. x-tile (16x2048 f32) + B-tile live in
//    LDS (320KB/WGP) for all 50 ISTA iterations -> zero HBM traffic in loop.
//  * conv1d(K=100) expressed as 16x116x16 GEMM with a constant banded
//    "kernel matrix" held in registers -> V_WMMA_F32_16X16X4_F32 inner loop
//    (29 WMMAs per 16x16 output tile), exact fp32 like the reference.
//  * Final x@W1 (+bias, relu) uses the same f32 WMMA path; 256->1 head is a
//    per-lane dot product.
//
// WMMA VGPR layout assumptions (ISA 7.12.2, compile-only, not runtime-checked):
//  * A (16x4 f32, 2 VGPRs): lane l<16: {v0,v1} = A[M=l][K=0,1];
//                           lane l>=16: {v0,v1} = A[M=l-16][K=2,3].
//  * B (4x16 f32, 2 VGPRs): mirrored: lanes 0-15 rows K=0,1 ; lanes16-31 K=2,3.
//  * C/D (16x16 f32, 8 VGPRs): element (M = vi + 8*(lane>>4), N = lane&15).
// ============================================================================

typedef float v2f __attribute__((ext_vector_type(2)));
typedef float v8f __attribute__((ext_vector_type(8)));

#define SIGLEN   2048
#define PULSE    100
#define ITERS    50
#define HIDDEN   256
#define BATCHN   4096
#define LAMV     0.1f
#define PADL     49            // (K-1)/2
#define KCHUNKS  29            // ceil((100+15)/4) -> K-dim = 116 (band rows >=100-? are 0)
#define ROWS     16            // batch rows per workgroup
#define XS_STRIDE 2164         // 49 halo + 2048 + 67 halo (%64 != 0 -> no bank conflict)
#define BS_STRIDE 2052         // %64 != 0
#define NTHREADS 512           // 16 waves (wave32)
#define NWAVES   16
#define TPW      8             // 128 position tiles / 16 waves

__global__ __launch_bounds__(NTHREADS) void ista_wmma_kernel(
    const float* __restrict__ y, const float* __restrict__ pulse,
    const float* __restrict__ W1, const float* __restrict__ b1,
    const float* __restrict__ W2, const float* __restrict__ b2,
    float* __restrict__ out_x, float* __restrict__ out_peak)
{
    extern __shared__ float lds[];
    float* xs = lds;                         // [ROWS][XS_STRIDE], padded coords: p = n + 49
    float* bs = lds + ROWS * XS_STRIDE;      // [ROWS][BS_STRIDE]

    const int tid  = threadIdx.x;
    const int wave = tid >> 5;
    const int lane = tid & 31;
    const int h    = lane >> 4;              // lane half
    const int ln   = lane & 15;
    const int row0 = blockIdx.x * ROWS;

    // ---- L = sum(pulse^2), theta = LAM/L (uniform -> scalar loads) ----
    float L = 0.f;
    for (int i = 0; i < PULSE; ++i) { float p = pulse[i]; L += p * p; }
    const float invL  = 1.0f / L;
    const float theta = LAMV * invL;

    // ---- banded kernel matrix fragments, register resident (58 VGPRs) ----
    // Band[j][s] = kern[j-s] (0 outside), kern[k] = pulse[99-k]; j = 4c+2h+{0,1}, s = ln
    v2f band[KCHUNKS];
#pragma unroll
    for (int c = 0; c < KCHUNKS; ++c) {
        int j0 = 4 * c + 2 * h;
        int k0 = j0 - ln, k1 = j0 + 1 - ln;
        band[c].x = (k0 >= 0 && k0 < PULSE) ? pulse[PULSE - 1 - k0] : 0.f;
        band[c].y = (k1 >= 0 && k1 < PULSE) ? pulse[PULSE - 1 - k1] : 0.f;
    }

    // ---- load y tile into xs with zeroed halo ----
    for (int idx = tid; idx < ROWS * XS_STRIDE; idx += NTHREADS) {
        int r = idx / XS_STRIDE, p = idx - r * XS_STRIDE;
        int n = p - PADL;
        xs[idx] = (n >= 0 && n < SIGLEN) ? y[(size_t)(row0 + r) * SIGLEN + n] : 0.f;
    }
    __syncthreads();

    v8f acc[TPW];

    // ---- B = conv(y, kernel)/L via WMMA ----
#pragma unroll
    for (int tt = 0; tt < TPW; ++tt) {
        int n0 = (wave * TPW + tt) * 16;
        v8f a = {};
#pragma unroll
        for (int c = 0; c < KCHUNKS; ++c) {
            v2f av = *(const v2f*)&xs[ln * XS_STRIDE + n0 + 4 * c + 2 * h]; // ds_load_b64
            a = __builtin_amdgcn_wmma_f32_16x16x4_f32(false, av, false, band[c],
                                                      (short)0, a, false, false);
        }
        acc[tt] = a;
    }
#pragma unroll
    for (int tt = 0; tt < TPW; ++tt) {
        int n0 = (wave * TPW + tt) * 16;
#pragma unroll
        for (int vi = 0; vi < 8; ++vi)
            bs[(vi + 8 * h) * BS_STRIDE + n0 + ln] = acc[tt][vi] * invL;
    }
    __syncthreads();

    // ---- x0 = 0 (halo stays zero forever) ----
    for (int idx = tid; idx < ROWS * XS_STRIDE; idx += NTHREADS) xs[idx] = 0.f;
    __syncthreads();

    // ---- 50 ISTA iterations, fully LDS-resident ----
    for (int it = 0; it < ITERS; ++it) {
        // conv(x) for all 8 owned tiles -> registers (all xs reads before any write)
#pragma unroll
        for (int tt = 0; tt < TPW; ++tt) {
            int n0 = (wave * TPW + tt) * 16;
            v8f a = {};
#pragma unroll
            for (int c = 0; c < KCHUNKS; ++c) {
                v2f av = *(const v2f*)&xs[ln * XS_STRIDE + n0 + 4 * c + 2 * h];
                a = __builtin_amdgcn_wmma_f32_16x16x4_f32(false, av, false, band[c],
                                                          (short)0, a, false, false);
            }
            acc[tt] = a;
        }
        __syncthreads();   // every wave finished reading xs (incl. halo of neighbors)

        // c = B + x - conv(x)/L ; x = sign(c)*max(|c|-theta, 0)
#pragma unroll
        for (int tt = 0; tt < TPW; ++tt) {
            int n0 = (wave * TPW + tt) * 16;
#pragma unroll
            for (int vi = 0; vi < 8; ++vi) {
                int m = vi + 8 * h;
                float xv = xs[m * XS_STRIDE + PADL + n0 + ln];   // own region only
                float bv = bs[m * BS_STRIDE + n0 + ln];
                float cv = bv + xv - acc[tt][vi] * invL;
                float mag = fabsf(cv) - theta;
                mag = mag > 0.f ? mag : 0.f;
                xs[m * XS_STRIDE + PADL + n0 + ln] = copysignf(mag, cv);
            }
        }
        __syncthreads();   // writes visible before next iteration's reads
    }

    // ---- emit x ----
    for (int idx = tid; idx < ROWS * SIGLEN; idx += NTHREADS) {
        int r = idx >> 11, n = idx & (SIGLEN - 1);
        out_x[(size_t)(row0 + r) * SIGLEN + n] = xs[r * XS_STRIDE + PADL + n];
    }
    __syncthreads();

    // ---- features = relu(x @ W1 + b1): wave w owns output cols [16w,16w+16) ----
    float* fs = bs;   // reuse B region: [ROWS][HIDDEN]
    if (wave < NWAVES) {
        int col0 = wave * 16;
        v8f a = {};
        for (int kc = 0; kc < SIGLEN / 4; ++kc) {
            int kbase = 4 * kc + 2 * h;
            v2f av = *(const v2f*)&xs[ln * XS_STRIDE + PADL + kbase];
            v2f bvf;
            bvf.x = W1[(size_t)kbase * HIDDEN + col0 + ln];
            bvf.y = W1[(size_t)(kbase + 1) * HIDDEN + col0 + ln];
            a = __builtin_amdgcn_wmma_f32_16x16x4_f32(false, av, false, bvf,
                                                      (short)0, a, false, false);
        }
        float bias = b1[col0 + ln];
#pragma unroll
        for (int vi = 0; vi < 8; ++vi) {
            float f = a[vi] + bias;
            fs[(vi + 8 * h) * HIDDEN + col0 + ln] = f > 0.f ? f : 0.f;
        }
    }
    __syncthreads();

    // ---- peak_pred = features @ W2 + b2 ----
    if (wave == 0 && lane < ROWS) {
        float s = b2[0];
        for (int j = 0; j < HIDDEN; ++j) s += fs[lane * HIDDEN + j] * W2[j];
        out_peak[row0 + lane] = s;
    }
}

extern "C" void kernel_launch(void* const* d_in, const int* in_sizes, int n_in,
                              void* d_out, int out_size, void* d_ws, size_t ws_size,
                              hipStream_t stream) {
    const float* y     = (const float*)d_in[0];
    const float* pulse = (const float*)d_in[1];
    const float* W1    = (const float*)d_in[2];
    const float* b1    = (const float*)d_in[3];
    const float* W2    = (const float*)d_in[4];
    const float* b2    = (const float*)d_in[5];

    float* out_x    = (float*)d_out;
    float* out_peak = (float*)d_out + (size_t)BATCHN * SIGLEN;

    const size_t shmem = (size_t)(ROWS * XS_STRIDE + ROWS * BS_STRIDE) * sizeof(float);
    static_assert((ROWS * XS_STRIDE + ROWS * BS_STRIDE) * sizeof(float) <= 320 * 1024,
                  "LDS budget exceeded");
    hipFuncSetAttribute((const void*)ista_wmma_kernel,
                        hipFuncAttributeMaxDynamicSharedMemorySize, (int)shmem);

    dim3 grid(BATCHN / ROWS);   // 256 workgroups, one per 16 batch rows
    dim3 block(NTHREADS);       // 16 wave32s
    ista_wmma_kernel<<<grid, block, shmem, stream>>>(y, pulse, W1, b1, W2, b2,
                                                     out_x, out_peak);
}